// embedding_network_89567247991232
// MI455X (gfx1250) — compile-verified
//
#include <hip/hip_runtime.h>

#define NNODES 100000
#define NEDGES 1600000
#define DIM 64

typedef float v2f __attribute__((ext_vector_type(2)));
typedef float v8f __attribute__((ext_vector_type(8)));

// ---- zero deg + compute u = W3 @ relu(W4[:,0]) ----
__global__ void k_init(const float* __restrict__ W3, const float* __restrict__ W4,
                       float* __restrict__ deg, float* __restrict__ u, int n) {
  int i = blockIdx.x * 256 + threadIdx.x;
  if (i < n) deg[i] = 0.0f;
  if (i < DIM) {
    float s = 0.0f;
#pragma unroll 8
    for (int k = 0; k < DIM; ++k) {
      float w4 = W4[k];
      s += W3[i * DIM + k] * (w4 > 0.0f ? w4 : 0.0f);
    }
    u[i] = s;
  }
}

// ---- degree histogram ----
__global__ void k_degree(const int* __restrict__ rows, float* __restrict__ deg) {
  int e = blockIdx.x * 256 + threadIdx.x;
  if (e < NEDGES) atomicAdd(&deg[rows[e]], 1.0f);
}

// ---- zero a buffer (float4 granularity) ----
__global__ void k_zero4(float* __restrict__ p, int n4) {
  int i = blockIdx.x * 256 + threadIdx.x;
  if (i < n4) ((float4*)p)[i] = make_float4(0.f, 0.f, 0.f, 0.f);
}

// ---- scatter: neigh[row[e]] += emb[col[e]]  (1 wave per edge, 2 dims/lane) ----
__global__ void k_scatter(const int* __restrict__ rows, const int* __restrict__ cols,
                          const float* __restrict__ emb, float* __restrict__ neigh) {
  int idx = blockIdx.x * 256 + threadIdx.x;   // NEDGES*32 threads total
  int e = idx >> 5;
  int lane = idx & 31;
  if (e < NEDGES) {
    long long c = cols[e];
    long long r = rows[e];
    float2 v = *(const float2*)(emb + c * DIM + lane * 2);
    atomicAdd(&neigh[r * DIM + lane * 2 + 0], v.x);
    atomicAdd(&neigh[r * DIM + lane * 2 + 1], v.y);
  }
}

// ---- out = act(A @ W^T + addend)
// MODE 0: addend = deg⊗u, writes linear result to out_lin and relu to out_relu
// MODE 1: addend = base (full matrix), writes relu to out_relu
// 256 threads = 8 waves; wave w -> rows [blk*128 + w*16, +16), all 64 cols.
template <int MODE>
__global__ __launch_bounds__(256, 2)
void k_gemm64(const float* __restrict__ A, const float* __restrict__ W,
              const float* __restrict__ deg, const float* __restrict__ u,
              const float* __restrict__ addend,
              float* __restrict__ out_lin, float* __restrict__ out_relu,
              int n) {
  __shared__ float sW[DIM * 66];  // padded pitch: bank = (2*row + col) % 64
  __shared__ float su[DIM];

  int tid = threadIdx.x;
  for (int i = tid; i < DIM * DIM; i += 256)
    sW[(i >> 6) * 66 + (i & 63)] = W[i];
  if (MODE == 0 && tid < DIM) su[tid] = u[tid];
  __syncthreads();

  int wave = tid >> 5;
  int lane = tid & 31;
  int half = lane >> 4;   // 0: K-pair {0,1}; 1: K-pair {2,3} within each K=4 step
  int l16  = lane & 15;
  int row_base = blockIdx.x * 128 + wave * 16;

  // A-fragment source row (clamped so WMMA runs with full EXEC on dead tiles)
  int arow = row_base + l16;
  if (arow >= n) arow = n - 1;
  const float* Arow = A + (long long)arow * DIM;

  v8f acc[4];
#pragma unroll
  for (int t = 0; t < 4; ++t) acc[t] = (v8f){0.f, 0.f, 0.f, 0.f, 0.f, 0.f, 0.f, 0.f};

#pragma unroll 4
  for (int kk = 0; kk < 16; ++kk) {
    int kbase = kk * 4 + half * 2;
    v2f a = *(const v2f*)(Arow + kbase);          // A[m][kbase..kbase+1]
#pragma unroll
    for (int t = 0; t < 4; ++t) {
      // B = W^T : lane holds column N=l16 of B => row (t*16+l16) of W
      v2f b = *(const v2f*)(&sW[(t * 16 + l16) * 66 + kbase]);
      acc[t] = __builtin_amdgcn_wmma_f32_16x16x4_f32(
          /*neg_a=*/false, a, /*neg_b=*/false, b,
          /*c_mod=*/(short)0, acc[t], /*reuse_a=*/false, /*reuse_b=*/false);
    }
  }

  // Wave-uniform tail guard: row tiles are 16-aligned, so a tile is either
  // fully in-range or fully out (n may be any value; partial tiles only occur
  // when row_base + 16 > n, in which case rows >= n exist only beyond n-1 and
  // we simply skip — with n % 16 == 0 as here, no partial tiles at all).
  if (row_base + 16 <= n) {
    // D layout: VGPR r -> rows {r, r+8} across lane halves, col = l16 + 16*t
#pragma unroll
    for (int r = 0; r < 8; ++r) {
      int row = row_base + r + half * 8;
      long long rbase = (long long)row * DIM;
      float dv = (MODE == 0) ? deg[row] : 0.0f;
#pragma unroll
      for (int t = 0; t < 4; ++t) {
        int col = t * 16 + l16;
        float v = acc[t][r];
        if (MODE == 0) {
          v += dv * su[col];
          out_lin[rbase + col] = v;       // base (pre-activation)
        } else {
          v += addend[rbase + col];       // + base
        }
        out_relu[rbase + col] = v > 0.0f ? v : 0.0f;
      }
    }
  }
}

extern "C" void kernel_launch(void* const* d_in, const int* in_sizes, int n_in,
                              void* d_out, int out_size, void* d_ws, size_t ws_size,
                              hipStream_t stream) {
  const float* Xv  = (const float*)d_in[0];
  const int*  edge = (const int*)d_in[1];
  const float* W1  = (const float*)d_in[2];
  const float* W2  = (const float*)d_in[3];
  const float* W3  = (const float*)d_in[4];
  const float* W4  = (const float*)d_in[5];
  float* out = (float*)d_out;             // 100000 x 64, doubles as emb buffer

  float* ws    = (float*)d_ws;
  float* deg   = ws;                                      // NNODES
  float* u     = ws + ((NNODES + 255) & ~255);            // 64
  float* base  = u + 256;                                 // NNODES*64
  float* neigh = base + (size_t)NNODES * DIM;             // NNODES*64

  const int* rows = edge;
  const int* cols = edge + NEDGES;

  dim3 blk(256);
  k_init<<<(NNODES + 255) / 256, blk, 0, stream>>>(W3, W4, deg, u, NNODES);
  k_degree<<<(NEDGES + 255) / 256, blk, 0, stream>>>(rows, deg);

  // t = 1: emb = relu(base); also stash linear base for later steps
  k_gemm64<0><<<(NNODES + 127) / 128, blk, 0, stream>>>(
      Xv, W1, deg, u, nullptr, base, out, NNODES);

  // t = 2..4: neigh = scatter(emb); emb = relu(base + neigh @ W2^T)
  for (int t = 1; t < 4; ++t) {
    k_zero4<<<(NNODES * DIM / 4 + 255) / 256, blk, 0, stream>>>(neigh, NNODES * DIM / 4);
    k_scatter<<<(NEDGES * 32) / 256, blk, 0, stream>>>(rows, cols, out, neigh);
    k_gemm64<1><<<(NNODES + 127) / 128, blk, 0, stream>>>(
        neigh, W2, nullptr, nullptr, base, nullptr, out, NNODES);
  }
}